// GraphConvolutionalNetwork_74010876444913
// MI455X (gfx1250) — compile-verified
//
#include <hip/hip_runtime.h>
#include <stdint.h>

typedef __attribute__((ext_vector_type(2))) float v2f;
typedef __attribute__((ext_vector_type(8))) float v8f;

#define IN_F  128
#define HID   128
#define OUT_F 64
#define APAD  4   // floats of padding per LDS A-row (breaks 512B bank stride)

// ---------------- degree / normalization ----------------

__global__ void k_deg_init(float* __restrict__ deg, int n) {
  int i = blockIdx.x * blockDim.x + threadIdx.x;
  if (i < n) deg[i] = 1.0f;  // self-loop weight
}

__global__ void k_deg_accum(const long long* __restrict__ dst,
                            const float* __restrict__ w,
                            float* __restrict__ deg, int E) {
  int e = blockIdx.x * blockDim.x + threadIdx.x;
  if (e < E) unsafeAtomicAdd(&deg[(int)dst[e]], w[e]);   // hw fp32 atomic, resolved in L2
}

__global__ void k_dinv(const float* __restrict__ deg, float* __restrict__ dinv, int n) {
  int i = blockIdx.x * blockDim.x + threadIdx.x;
  if (i < n) {
    float d = deg[i];
    dinv[i] = d > 0.0f ? rsqrtf(d) : 0.0f;
  }
}

// -------- async global -> LDS copy helper (16B per lane, ASYNCcnt-tracked) --------
__device__ __forceinline__ void async_copy16(const void* gsrc, unsigned lds_addr) {
  unsigned long long ga = (unsigned long long)(uintptr_t)gsrc;
  asm volatile("global_load_async_to_lds_b128 %0, %1, off"
               :: "v"(lds_addr), "v"(ga) : "memory");
}

// ---------------- fp32 WMMA GEMM: Y[M,N] = act(X[M,K]) @ W[K,N] ----------------
// One block per 16-row strip; blockDim = 32*(N/16); wave w owns column tile w.
// A strip (16 x K) and W (K x N) are staged to LDS via global_load_async_to_lds_b128,
// then V_WMMA_F32_16X16X4_F32 consumes them from LDS (ds_load).
// A 16x4 f32 layout (ISA 7.12.2): lanes 0-15 hold M rows {K=k,k+1}, lanes 16-31 same rows {K=k+2,k+3}.
template <bool RELU_A>
__global__ void k_gemm_wmma_lds(const float* __restrict__ X, const float* __restrict__ W,
                                float* __restrict__ Y, int K, int N) {
  extern __shared__ char smem[];
  float* sA = (float*)smem;                       // 16 x (K+APAD)
  const int aFloats = 16 * (K + APAD);
  float* sW = sA + aFloats;                       // K x N

  const int tid = threadIdx.x;
  const int tm  = blockIdx.x << 4;                // row strip base

  // stage A strip: 16 rows of K floats, padded rows in LDS
  {
    const int qpr = K >> 2;                       // 16B chunks per row
    const int chunksA = 16 * qpr;
    for (int c = tid; c < chunksA; c += blockDim.x) {
      int row = c / qpr, q = c - row * qpr;
      const float* g = X + (size_t)(tm + row) * K + (q << 2);
      unsigned lds = (unsigned)(uintptr_t)(sA + row * (K + APAD) + (q << 2));
      async_copy16(g, lds);
    }
  }
  // stage W: flat contiguous K*N floats
  {
    const int chunksW = (K * N) >> 2;
    for (int c = tid; c < chunksW; c += blockDim.x) {
      const float* g = W + ((size_t)c << 2);
      unsigned lds = (unsigned)(uintptr_t)(sW + (c << 2));
      async_copy16(g, lds);
    }
  }
  asm volatile("s_wait_asynccnt 0x0" ::: "memory");
  __syncthreads();

  // compute: one 16x16 tile per wave
  const int lane = tid & 31;
  const int wv   = tid >> 5;
  const int half = lane >> 4;                     // 0: K=k,k+1  1: K=k+2,k+3
  const int l    = lane & 15;
  const int col  = (wv << 4) + l;
  const float* arow = sA + l * (K + APAD);

  v8f c = {};
  for (int k = 0; k < K; k += 4) {
    const int ka = k + (half << 1);
    v2f a, b;
    a.x = arow[ka];
    a.y = arow[ka + 1];
    if (RELU_A) { a.x = fmaxf(a.x, 0.0f); a.y = fmaxf(a.y, 0.0f); }
    b.x = sW[ka * N + col];
    b.y = sW[(ka + 1) * N + col];
    c = __builtin_amdgcn_wmma_f32_16x16x4_f32(false, a, false, b, (short)0, c, false, false);
  }

  float* y = Y + (size_t)(tm + (half << 3)) * N + col;
#pragma unroll
  for (int r = 0; r < 8; ++r) y[(size_t)r * N] = c[r];
}

// ---------------- self-loop + bias init: out[i,f] = b[f] + dinv[i]^2 * XW[i,f] ----------------

__global__ void k_self_bias(const float* __restrict__ XW, const float* __restrict__ dinv,
                            const float* __restrict__ bias, float* __restrict__ out,
                            int n, int F) {
  int idx = blockIdx.x * blockDim.x + threadIdx.x;
  if (idx >= n * F) return;
  int i = idx / F;
  int f = idx - i * F;
  float di = dinv[i];
  out[idx] = bias[f] + di * di * XW[idx];
}

// ---------------- edge scatter: out[dst] += XW[src] * (dinv[src]*w*dinv[dst]) ----------------
// One wave per edge; each lane handles F/32 consecutive features (4 for F=128, 2 for F=64).
__global__ void k_edge_scatter(const long long* __restrict__ src,
                               const long long* __restrict__ dst,
                               const float* __restrict__ w,
                               const float* __restrict__ dinv,
                               const float* __restrict__ XW,
                               float* __restrict__ out, int E, int F) {
  int gid  = blockIdx.x * blockDim.x + threadIdx.x;
  int e    = gid >> 5;
  if (e >= E) return;
  int lane = gid & 31;

  int   s    = (int)src[e];
  int   d    = (int)dst[e];
  float norm = dinv[s] * w[e] * dinv[d];

  const int per = F >> 5;
  const float* xs = XW  + (size_t)s * F + lane * per;
  float*       o  = out + (size_t)d * F + lane * per;
#pragma unroll 4
  for (int j = 0; j < per; ++j)
    unsafeAtomicAdd(&o[j], xs[j] * norm);
}

// ---------------- launcher ----------------

extern "C" void kernel_launch(void* const* d_in, const int* in_sizes, int n_in,
                              void* d_out, int out_size, void* d_ws, size_t ws_size,
                              hipStream_t stream) {
  const float*     x   = (const float*)d_in[0];
  const long long* ei  = (const long long*)d_in[1];   // int64 edge_index [2, E]
  const float*     ew  = (const float*)d_in[2];
  const float*     W1  = (const float*)d_in[3];
  const float*     b1  = (const float*)d_in[4];
  const float*     W2  = (const float*)d_in[5];
  const float*     b2  = (const float*)d_in[6];
  float*           out = (float*)d_out;

  const int N = in_sizes[0] / IN_F;
  const int E = in_sizes[2];
  const long long* src = ei;
  const long long* dst = ei + E;

  // workspace layout (all fits in the 192MB L2): deg | dinv | XW1 | h | HW2
  float* deg  = (float*)d_ws;                 // N
  float* dinv = deg  + N;                     // N
  float* XW1  = dinv + N;                     // N*HID
  float* h    = XW1  + (size_t)N * HID;       // N*HID
  float* HW2  = h    + (size_t)N * HID;       // N*OUT_F

  const int B = 256;  // 8 wave32 waves per block

  // normalization (shared by both layers)
  k_deg_init <<<(N + B - 1) / B, B, 0, stream>>>(deg, N);
  k_deg_accum<<<(E + B - 1) / B, B, 0, stream>>>(dst, ew, deg, E);
  k_dinv     <<<(N + B - 1) / B, B, 0, stream>>>(deg, dinv, N);

  const int strips = N >> 4;  // N = 100000 -> 6250 row strips, exact

  // layer 1: XW1 = x @ W1 ; h = b1 + dinv^2*XW1 ; scatter edges into h
  {
    size_t shmem = (size_t)(16 * (IN_F + APAD) + IN_F * HID) * sizeof(float);  // ~74KB (WGP has 320KB)
    k_gemm_wmma_lds<false><<<strips, 32 * (HID / 16), shmem, stream>>>(x, W1, XW1, IN_F, HID);
    k_self_bias<<<((N * HID) + B - 1) / B, B, 0, stream>>>(XW1, dinv, b1, h, N, HID);
    unsigned long long th = (unsigned long long)E * 32ull;
    k_edge_scatter<<<(unsigned)((th + B - 1) / B), B, 0, stream>>>(src, dst, ew, dinv, XW1, h, E, HID);
  }

  // layer 2: HW2 = relu(h) @ W2 (relu folded into LDS A-read) ; out = b2 + dinv^2*HW2 ; scatter
  {
    size_t shmem = (size_t)(16 * (HID + APAD) + HID * OUT_F) * sizeof(float);  // ~41KB
    k_gemm_wmma_lds<true><<<strips, 32 * (OUT_F / 16), shmem, stream>>>(h, W2, HW2, HID, OUT_F);
    k_self_bias<<<((N * OUT_F) + B - 1) / B, B, 0, stream>>>(HW2, dinv, b2, out, N, OUT_F);
    unsigned long long th = (unsigned long long)E * 32ull;
    k_edge_scatter<<<(unsigned)((th + B - 1) / B), B, 0, stream>>>(src, dst, ew, dinv, HW2, out, E, OUT_F);
  }
}